// GCNEncoder_21406117003578
// MI455X (gfx1250) — compile-verified
//
#include <hip/hip_runtime.h>
#include <hip/hip_bf16.h>

// GCN encoder for MI455X (gfx1250, wave32).
//   layer: feat = X @ W^T (WMMA f32 16x16x4)  ->  agg = self + sum_e norm*feat[src]
//          h = relu(agg + b)
// Aggregation is L2-resident (25.6MB feature matrix << 192MB L2), done with
// native f32 global atomics. GEMMs use V_WMMA_F32_16X16X4_F32 for exact f32.

typedef __attribute__((ext_vector_type(2))) float v2f;
typedef __attribute__((ext_vector_type(8))) float v8f;

#define NNODES 50000
#define CH 128          // in/hidden channels
#define OUTC 64

// ---------------- degree / normalization ----------------

__global__ __launch_bounds__(256) void k_deg_init(unsigned int* deg, int n) {
  int i = blockIdx.x * 256 + threadIdx.x;
  if (i < n) deg[i] = 1u;                       // self-loop
}

__global__ __launch_bounds__(256) void k_deg_count(unsigned int* deg,
                                                   const int* __restrict__ dst, int e) {
  int i = blockIdx.x * 256 + threadIdx.x;
  if (i < e) {
    __hip_atomic_fetch_add(&deg[dst[i]], 1u, __ATOMIC_RELAXED, __HIP_MEMORY_SCOPE_AGENT);
  }
}

__global__ __launch_bounds__(256) void k_deg_fin(float* dinv_deg, int n) {
  int i = blockIdx.x * 256 + threadIdx.x;
  if (i < n) {
    unsigned int d = ((const unsigned int*)dinv_deg)[i];
    dinv_deg[i] = (d > 0u) ? rsqrtf((float)d) : 0.0f;   // in-place uint -> float
  }
}

// ---------------- dense: Y[N,out_c] = X[N,128] @ W[out_c,128]^T (+bias,relu) ---

__global__ __launch_bounds__(256) void k_gemm_wmma(const float* __restrict__ X,
                                                   const float* __restrict__ W,
                                                   const float* __restrict__ bias,
                                                   float* __restrict__ Y,
                                                   int out_c, int do_relu) {
  __shared__ float lW[CH * CH];                 // 64 KB: whole weight matrix
  const int tid = threadIdx.x;
  const int m0 = blockIdx.x * 16;               // 16 rows per block (N % 16 == 0)

  // cooperative load of W into LDS (out_c*128 floats, float4 vectorized)
  {
    const float4* wg = (const float4*)W;
    float4* ws = (float4*)lW;
    const int wtot = out_c * (CH / 4);
    for (int i = tid; i < wtot; i += 256) ws[i] = wg[i];
  }
  __syncthreads();

  const int wave = tid >> 5;
  const int lane = tid & 31;
  const int o0 = wave * 16;                     // 16 output channels per wave
  if (o0 < out_c) {                             // uniform per wave -> EXEC all-1s
    const int half = lane >> 4;                 // 0: lanes 0-15, 1: lanes 16-31
    const int l15 = lane & 15;
    const float* xrow = X + (size_t)(m0 + l15) * CH;
    const float* wrow = lW + (o0 + l15) * CH;

    v8f acc = {0.f, 0.f, 0.f, 0.f, 0.f, 0.f, 0.f, 0.f};
#pragma unroll 8
    for (int k0 = 0; k0 < CH; k0 += 4) {
      const int ka = k0 + half * 2;             // lane half selects K pair
      v2f a = *(const v2f*)(xrow + ka);         // A[16x4]: row=l15, K=ka,ka+1
      v2f b = *(const v2f*)(wrow + ka);         // B[4x16]: col=l15, K=ka,ka+1
      acc = __builtin_amdgcn_wmma_f32_16x16x4_f32(
          false, a, false, b, (short)0, acc, false, false);
    }

    // D layout: VGPR i -> row m0+i (lanes 0-15) / m0+8+i (lanes 16-31), col=o0+l15
    const int col = o0 + l15;
    const int rbase = m0 + half * 8;
    const float bv = bias ? bias[col] : 0.0f;
#pragma unroll
    for (int i = 0; i < 8; ++i) {
      float v = acc[i] + bv;
      if (do_relu) v = fmaxf(v, 0.0f);
      Y[(size_t)(rbase + i) * out_c + col] = v;
    }
  }
}

// ---------------- aggregation ----------------

// agg = feat * dinv^2  (self-loop term; also zero-fills accumulator)
__global__ __launch_bounds__(256) void k_self_init(const float4* __restrict__ feat,
                                                   const float* __restrict__ dinv,
                                                   float4* __restrict__ agg, int n) {
  int t = blockIdx.x * 256 + threadIdx.x;       // n * 32 float4 slots
  if (t < n * (CH / 4)) {
    int node = t >> 5;
    float w = dinv[node];
    w *= w;
    float4 v = feat[t];
    agg[t] = make_float4(v.x * w, v.y * w, v.z * w, v.w * w);
  }
}

// one wave per edge, each lane scatters 4 channels via f32 atomics (L2-resident)
__global__ __launch_bounds__(256) void k_edge_agg(const float4* __restrict__ feat,
                                                  float* __restrict__ out,
                                                  const int* __restrict__ src,
                                                  const int* __restrict__ dst,
                                                  const float* __restrict__ dinv, int e) {
  int w = (blockIdx.x * 256 + threadIdx.x) >> 5;
  int lane = threadIdx.x & 31;
  if (w >= e) return;
  int s = src[w];
  int d = dst[w];
  float norm = dinv[s] * dinv[d];
  float4 v = feat[(size_t)s * (CH / 4) + lane];
  float* o = out + (size_t)d * CH + lane * 4;
  __hip_atomic_fetch_add(o + 0, v.x * norm, __ATOMIC_RELAXED, __HIP_MEMORY_SCOPE_AGENT);
  __hip_atomic_fetch_add(o + 1, v.y * norm, __ATOMIC_RELAXED, __HIP_MEMORY_SCOPE_AGENT);
  __hip_atomic_fetch_add(o + 2, v.z * norm, __ATOMIC_RELAXED, __HIP_MEMORY_SCOPE_AGENT);
  __hip_atomic_fetch_add(o + 3, v.w * norm, __ATOMIC_RELAXED, __HIP_MEMORY_SCOPE_AGENT);
}

// h = relu(agg + b)
__global__ __launch_bounds__(256) void k_bias_relu(const float4* __restrict__ agg,
                                                   const float* __restrict__ b,
                                                   float4* __restrict__ h, int n) {
  int t = blockIdx.x * 256 + threadIdx.x;
  if (t < n * (CH / 4)) {
    float4 bv = ((const float4*)b)[t & 31];
    float4 v = agg[t];
    h[t] = make_float4(fmaxf(v.x + bv.x, 0.f), fmaxf(v.y + bv.y, 0.f),
                       fmaxf(v.z + bv.z, 0.f), fmaxf(v.w + bv.w, 0.f));
  }
}

// ---------------- launch ----------------

extern "C" void kernel_launch(void* const* d_in, const int* in_sizes, int n_in,
                              void* d_out, int out_size, void* d_ws, size_t ws_size,
                              hipStream_t stream) {
  const float* x  = (const float*)d_in[0];
  const int*   ei = (const int*)d_in[1];        // [2,E] int32 (harness dtype rule)
  const float* W1 = (const float*)d_in[2];
  const float* b1 = (const float*)d_in[3];
  const float* W2 = (const float*)d_in[4];
  const float* b2 = (const float*)d_in[5];
  const float* Wp = (const float*)d_in[6];
  const float* bp = (const float*)d_in[7];

  const int N = in_sizes[0] / CH;               // 50000
  const int E = in_sizes[1] / 2;                // 800000
  const int* src = ei;
  const int* dst = ei + E;

  // workspace: dinv[N] | bufA[N*128] | bufB[N*128]  (~51.4 MB)
  float* dinv = (float*)d_ws;
  float* bufA = dinv + ((N + 63) & ~63);
  float* bufB = bufA + (size_t)N * CH;

  const int nBlk   = (N + 255) / 256;
  const int eBlk   = (E + 255) / 256;
  const int featBlk = (N * (CH / 4) + 255) / 256;   // N*32 float4 elems
  const int edgeWBlk = (E * 32 + 255) / 256;        // one wave per edge
  const int gemmBlk = N / 16;                       // 3125

  // normalization
  k_deg_init<<<nBlk, 256, 0, stream>>>((unsigned int*)dinv, N);
  k_deg_count<<<eBlk, 256, 0, stream>>>((unsigned int*)dinv, dst, E);
  k_deg_fin<<<nBlk, 256, 0, stream>>>(dinv, N);

  // layer 1: feat1=bufA, agg1=bufB, h1=bufA
  k_gemm_wmma<<<gemmBlk, 256, 0, stream>>>(x, W1, nullptr, bufA, CH, 0);
  k_self_init<<<featBlk, 256, 0, stream>>>((const float4*)bufA, dinv, (float4*)bufB, N);
  k_edge_agg<<<edgeWBlk, 256, 0, stream>>>((const float4*)bufA, bufB, src, dst, dinv, E);
  k_bias_relu<<<featBlk, 256, 0, stream>>>((const float4*)bufB, b1, (float4*)bufA, N);

  // layer 2: feat2=bufB, agg2=bufA, h2=bufB
  k_gemm_wmma<<<gemmBlk, 256, 0, stream>>>(bufA, W2, nullptr, bufB, CH, 0);
  k_self_init<<<featBlk, 256, 0, stream>>>((const float4*)bufB, dinv, (float4*)bufA, N);
  k_edge_agg<<<edgeWBlk, 256, 0, stream>>>((const float4*)bufB, bufA, src, dst, dinv, E);
  k_bias_relu<<<featBlk, 256, 0, stream>>>((const float4*)bufA, b2, (float4*)bufB, N);

  // projection: d_out = relu(h2 @ Wp^T + bp), [N,64]
  k_gemm_wmma<<<gemmBlk, 256, 0, stream>>>(bufB, Wp, bp, (float*)d_out, OUTC, 1);
}